// GCNLayer_37031208026784
// MI455X (gfx1250) — compile-verified
//
#include <hip/hip_runtime.h>

#define IN_FEAT 128
#define OUT_FEAT 64
#define LDS_STRIDE 80   // 64 + 16 pad: half-waves land on disjoint bank groups

typedef __attribute__((ext_vector_type(2))) float v2f;
typedef __attribute__((ext_vector_type(8))) float v8f;

// ---------------------------------------------------------------------------
// Stage 1: support = x @ weight  via V_WMMA_F32_16X16X4_F32 (full f32 precision)
// block = 128 threads = 4 waves; block tile = 16 rows x 64 cols;
// wave w computes the 16x16 tile at columns [16w, 16w+16).
// ---------------------------------------------------------------------------
__global__ __launch_bounds__(128) void gcn_gemm_wmma(
    const float* __restrict__ x, const float* __restrict__ w,
    float* __restrict__ support, int n_nodes) {
  __shared__ float wlds[IN_FEAT * LDS_STRIDE];

  const int tid = threadIdx.x;
  // Stage the 128x64 weight into LDS (padded stride 80 floats).
  for (int i = tid; i < IN_FEAT * OUT_FEAT; i += 128) {
    int r = i >> 6;        // k row
    int c = i & 63;        // out col
    wlds[r * LDS_STRIDE + c] = w[i];
  }
  __syncthreads();

  const int wave = tid >> 5;          // 0..3 -> column tile
  const int lane = tid & 31;
  const int half = lane >> 4;         // 0: lanes 0-15, 1: lanes 16-31
  const int l16  = lane & 15;
  const int row_base = blockIdx.x * 16;
  const int col_base = wave * 16;

  // A-operand row for this lane (lanes 0-15 and 16-31 both cover M=0..15).
  int arow = row_base + l16;
  if (arow >= n_nodes) arow = n_nodes - 1;   // clamp; no divergence before WMMA
  const float* __restrict__ xrow = x + (long long)arow * IN_FEAT;

  v8f acc = {};
#pragma unroll
  for (int k = 0; k < IN_FEAT; k += 4) {
    const int kk = k + half * 2;      // lanes 0-15: K={k,k+1}; 16-31: {k+2,k+3}
    // A 16x4 f32 layout (ISA 7.12.2): VGPR0=K=kk, VGPR1=K=kk+1 -> one b64 load
    v2f a = *(const v2f*)(xrow + kk);
    // B 4x16 f32 layout (symmetric): row K striped across lanes within a VGPR
    v2f b;
    b.x = wlds[(kk + 0) * LDS_STRIDE + col_base + l16];
    b.y = wlds[(kk + 1) * LDS_STRIDE + col_base + l16];
    acc = __builtin_amdgcn_wmma_f32_16x16x4_f32(
        /*neg_a=*/false, a, /*neg_b=*/false, b,
        /*c_mod=*/(short)0, acc, /*reuse_a=*/false, /*reuse_b=*/false);
  }

  // C/D layout: VGPR i -> M = i + 8*half, N = l16.
#pragma unroll
  for (int i = 0; i < 8; ++i) {
    const int orow = row_base + half * 8 + i;
    if (orow < n_nodes)
      support[(long long)orow * OUT_FEAT + col_base + l16] = acc[i];
  }
}

// ---------------------------------------------------------------------------
// Zero-init the output (harness poisons d_out with 0xAA).
// ---------------------------------------------------------------------------
__global__ __launch_bounds__(256) void gcn_zero(float* __restrict__ out, int n) {
  int i = blockIdx.x * blockDim.x + threadIdx.x;
  if (i < n) out[i] = 0.0f;
}

// ---------------------------------------------------------------------------
// Stage 2: out[r] += vals[e] * support[c]  (edge-parallel scatter).
// 16 lanes per edge, each lane owns a float4 (16B) slice of the 64 features.
// support (25.6 MB) and out (25.6 MB) are both L2-resident (192 MB L2), so
// this runs at L2 gather + L2 fp32-atomic throughput.
// ---------------------------------------------------------------------------
__global__ __launch_bounds__(256) void gcn_spmm(
    const float* __restrict__ support, const int* __restrict__ rows,
    const int* __restrict__ cols, const float* __restrict__ vals,
    float* __restrict__ out, int n_edges) {
  const long long t = (long long)blockIdx.x * blockDim.x + threadIdx.x;
  const long long e = t >> 4;
  if (e >= n_edges) return;
  const int f4 = (int)(t & 15) * 4;

  const int r = rows[e];
  const int c = cols[e];
  const float v = vals[e];

  const float4 s = *(const float4*)(support + (long long)c * OUT_FEAT + f4);
  float* o = out + (long long)r * OUT_FEAT + f4;
  unsafeAtomicAdd(o + 0, v * s.x);   // global_atomic_add_f32 (no-return)
  unsafeAtomicAdd(o + 1, v * s.y);
  unsafeAtomicAdd(o + 2, v * s.z);
  unsafeAtomicAdd(o + 3, v * s.w);
}

extern "C" void kernel_launch(void* const* d_in, const int* in_sizes, int n_in,
                              void* d_out, int out_size, void* d_ws, size_t ws_size,
                              hipStream_t stream) {
  const float* x    = (const float*)d_in[0];
  const float* w    = (const float*)d_in[1];
  const int*   rows = (const int*)d_in[2];
  const int*   cols = (const int*)d_in[3];
  const float* vals = (const float*)d_in[4];
  float* out     = (float*)d_out;
  float* support = (float*)d_ws;     // N * 64 floats = 25.6 MB scratch

  const int n_nodes = in_sizes[0] / IN_FEAT;
  const int n_edges = in_sizes[2];

  // 1) GEMM: one block per 16 node-rows.
  gcn_gemm_wmma<<<(n_nodes + 15) / 16, 128, 0, stream>>>(x, w, support, n_nodes);

  // 2) Zero the output.
  gcn_zero<<<(out_size + 255) / 256, 256, 0, stream>>>(out, out_size);

  // 3) Edge scatter: 16 threads per edge.
  const long long sp_threads = (long long)n_edges * 16;
  gcn_spmm<<<(unsigned)((sp_threads + 255) / 256), 256, 0, stream>>>(
      support, rows, cols, vals, out, n_edges);
}